// Encoder_75350906241750
// MI455X (gfx1250) — compile-verified
//
#include <hip/hip_runtime.h>
#include <math.h>

// ---------------- CDNA5 / gfx1250: wave32, WMMA f32 16x16x4 ----------------
typedef __attribute__((ext_vector_type(2))) float v2f;
typedef __attribute__((ext_vector_type(8))) float v8f;

#define FIN   128
#define HID   64
#define LAT   32
#define NGR   512
#define NEG_SLOPE 0.2f

// order-preserving float atomic max (works for mixed signs; m initialized to -inf)
__device__ __forceinline__ void atomicMaxF(float* addr, float val) {
    if (val >= 0.0f) atomicMax((int*)addr, __float_as_int(val));
    else             atomicMin((unsigned int*)addr, (unsigned int)__float_as_int(val));
}

// zero agg[n*HID], m=-inf, denom=0
__global__ void k_init(float* __restrict__ agg, float* __restrict__ m,
                       float* __restrict__ denom, int n) {
    int i = blockIdx.x * blockDim.x + threadIdx.x;
    if (i < n * HID) agg[i] = 0.0f;
    if (i < n) { m[i] = -__builtin_inff(); denom[i] = 0.0f; }
}

// H[n,HID] = X[n,K] @ W[K,HID]   (exact fp32 via V_WMMA_F32_16X16X4_F32)
// One wave computes a 16-row x 64-col tile; 4 v8f accumulators; K-loop step 4.
// A layout (16x4 f32): lane = (kh*16 + m): v2f = {A[m][k+2kh], A[m][k+2kh+1]}
// B layout (4x16 f32): lane = (kh*16 + c): v2f = {B[k+2kh][c], B[k+2kh+1][c]}
// C/D layout: lane = (kh*16 + c), vgpr v: D[row0 + 8*kh + v][c]
__global__ void k_gemm_wmma(const float* __restrict__ X, const float* __restrict__ W,
                            float* __restrict__ H, int n, int K) {
    __shared__ float Wt[FIN * HID];           // transposed: Wt[c*K + k] = W[k*HID + c]
    const int tid = threadIdx.x;
    for (int i = tid; i < K * HID; i += blockDim.x) {
        int k = i / HID, c = i % HID;
        Wt[c * K + k] = W[i];
    }
    __syncthreads();

    const int lane = tid & 31;
    const int wave = blockIdx.x * (blockDim.x >> 5) + (tid >> 5);
    const int row0 = wave * 16;
    const int m    = lane & 15;
    const int kh   = lane >> 4;               // 0 -> K pair (k,k+1); 1 -> (k+2,k+3)

    int r = row0 + m; if (r > n - 1) r = n - 1;        // clamp keeps EXEC all-ones
    const float* __restrict__ xrow = X + (size_t)r * K;
    const float* __restrict__ w0 = &Wt[(0 * 16 + m) * K];
    const float* __restrict__ w1 = &Wt[(1 * 16 + m) * K];
    const float* __restrict__ w2 = &Wt[(2 * 16 + m) * K];
    const float* __restrict__ w3 = &Wt[(3 * 16 + m) * K];

    v8f acc0 = {}, acc1 = {}, acc2 = {}, acc3 = {};
    for (int k = 0; k < K; k += 4) {
        const int ka = k + kh * 2;
        v2f a  = { xrow[ka], xrow[ka + 1] };
        v2f b0 = { w0[ka],   w0[ka + 1]   };
        v2f b1 = { w1[ka],   w1[ka + 1]   };
        v2f b2 = { w2[ka],   w2[ka + 1]   };
        v2f b3 = { w3[ka],   w3[ka + 1]   };
        acc0 = __builtin_amdgcn_wmma_f32_16x16x4_f32(false, a, false, b0, (short)0, acc0, false, false);
        acc1 = __builtin_amdgcn_wmma_f32_16x16x4_f32(false, a, false, b1, (short)0, acc1, false, false);
        acc2 = __builtin_amdgcn_wmma_f32_16x16x4_f32(false, a, false, b2, (short)0, acc2, false, false);
        acc3 = __builtin_amdgcn_wmma_f32_16x16x4_f32(false, a, false, b3, (short)0, acc3, false, false);
    }

    #pragma unroll
    for (int v = 0; v < 8; ++v) {
        int rr = row0 + kh * 8 + v;
        if (rr < n) {
            float* hp = H + (size_t)rr * HID + m;
            hp[0]  = acc0[v];
            hp[16] = acc1[v];
            hp[32] = acc2[v];
            hp[48] = acc3[v];
        }
    }
}

// per-node attention scores: es = h . a_src, ed = h . a_dst  (wave per node)
__global__ void k_score(const float* __restrict__ h, const float* __restrict__ a_src,
                        const float* __restrict__ a_dst, float* __restrict__ es,
                        float* __restrict__ ed, int n) {
    int node = (blockIdx.x * blockDim.x + threadIdx.x) >> 5;
    int lane = threadIdx.x & 31;
    if (node >= n) return;
    float s = 0.f, d = 0.f;
    #pragma unroll
    for (int f = 0; f < 2; ++f) {
        float hv = h[(size_t)node * HID + lane + 32 * f];
        s += hv * a_src[lane + 32 * f];
        d += hv * a_dst[lane + 32 * f];
    }
    #pragma unroll
    for (int off = 16; off > 0; off >>= 1) {
        s += __shfl_down(s, off, 32);
        d += __shfl_down(d, off, 32);
    }
    if (lane == 0) { es[node] = s; ed[node] = d; }
}

// raw score + LeakyReLU, segment max into m[dst]
__global__ void k_edge_max(const int* __restrict__ ei, int E, int n,
                           const float* __restrict__ es, const float* __restrict__ ed,
                           float* __restrict__ ebuf, float* __restrict__ m) {
    int e = blockIdx.x * blockDim.x + threadIdx.x;
    int ET = E + n;
    if (e >= ET) return;
    int s, d;
    if (e < E) { s = ei[e]; d = ei[E + e]; } else { s = d = e - E; }
    float v = es[s] + ed[d];
    v = (v > 0.f) ? v : NEG_SLOPE * v;
    ebuf[e] = v;
    atomicMaxF(&m[d], v);
}

// w = exp(e - m[dst]); segment sum into denom[dst]
__global__ void k_edge_expsum(const int* __restrict__ ei, int E, int n,
                              float* __restrict__ ebuf, const float* __restrict__ m,
                              float* __restrict__ denom) {
    int e = blockIdx.x * blockDim.x + threadIdx.x;
    int ET = E + n;
    if (e >= ET) return;
    int d = (e < E) ? ei[E + e] : (e - E);
    float w = __expf(ebuf[e] - m[d]);
    ebuf[e] = w;
    atomicAdd(&denom[d], w);
}

// agg[dst] += h[src] * (w / denom[dst])   (wave per edge, 2 features per lane)
__global__ void k_edge_agg(const int* __restrict__ ei, int E, int n,
                           const float* __restrict__ ebuf, const float* __restrict__ denom,
                           const float* __restrict__ h, float* __restrict__ agg) {
    int e    = (blockIdx.x * blockDim.x + threadIdx.x) >> 5;
    int lane = threadIdx.x & 31;
    int ET = E + n;
    if (e >= ET) return;
    int s, d;
    if (e < E) { s = ei[e]; d = ei[E + e]; } else { s = d = e - E; }
    float alpha = ebuf[e] / denom[d];
    atomicAdd(&agg[(size_t)d * HID + lane],      h[(size_t)s * HID + lane]      * alpha);
    atomicAdd(&agg[(size_t)d * HID + lane + 32], h[(size_t)s * HID + lane + 32] * alpha);
}

// x = relu(agg + b)  (in place)
__global__ void k_bias_relu(float* __restrict__ agg, const float* __restrict__ b, int n) {
    int i = blockIdx.x * blockDim.x + threadIdx.x;
    if (i >= n * HID) return;
    float v = agg[i] + b[i & (HID - 1)];
    agg[i] = v > 0.f ? v : 0.f;
}

__global__ void k_pool_init(float* __restrict__ sums, float* __restrict__ cnt) {
    int i = blockIdx.x * blockDim.x + threadIdx.x;
    if (i < NGR * HID) sums[i] = 0.f;
    if (i < NGR) cnt[i] = 0.f;
}

// wave per node: sums[batch[i]] += x[i]; cnt[batch[i]] += 1
__global__ void k_pool_acc(const float* __restrict__ x, const int* __restrict__ batch,
                           float* __restrict__ sums, float* __restrict__ cnt, int n) {
    int node = (blockIdx.x * blockDim.x + threadIdx.x) >> 5;
    int lane = threadIdx.x & 31;
    if (node >= n) return;
    int g = batch[node];
    atomicAdd(&sums[(size_t)g * HID + lane],      x[(size_t)node * HID + lane]);
    atomicAdd(&sums[(size_t)g * HID + lane + 32], x[(size_t)node * HID + lane + 32]);
    if (lane == 0) atomicAdd(&cnt[g], 1.0f);
}

// out[g,l] = (sums[g]/max(cnt,1)) . fcW[:,l] + fcb[l]
__global__ void k_fc(const float* __restrict__ sums, const float* __restrict__ cnt,
                     const float* __restrict__ fcW, const float* __restrict__ fcb,
                     float* __restrict__ out) {
    int i = blockIdx.x * blockDim.x + threadIdx.x;
    if (i >= NGR * LAT) return;
    int g = i / LAT, l = i % LAT;
    float c = cnt[g]; c = c > 1.f ? c : 1.f;
    float inv = 1.0f / c;
    float acc = 0.f;
    #pragma unroll
    for (int k = 0; k < HID; ++k) acc += sums[g * HID + k] * fcW[k * LAT + l];
    out[i] = acc * inv + fcb[l];
}

extern "C" void kernel_launch(void* const* d_in, const int* in_sizes, int n_in,
                              void* d_out, int out_size, void* d_ws, size_t ws_size,
                              hipStream_t stream) {
    const float* x    = (const float*)d_in[0];
    const int*   ei   = (const int*)  d_in[1];
    const int*   batch= (const int*)  d_in[2];
    const float* Wp[3]  = { (const float*)d_in[3], (const float*)d_in[7],  (const float*)d_in[11] };
    const float* asp[3] = { (const float*)d_in[4], (const float*)d_in[8],  (const float*)d_in[12] };
    const float* adp[3] = { (const float*)d_in[5], (const float*)d_in[9],  (const float*)d_in[13] };
    const float* bp[3]  = { (const float*)d_in[6], (const float*)d_in[10], (const float*)d_in[14] };
    const float* fcW = (const float*)d_in[15];
    const float* fcb = (const float*)d_in[16];
    float* out = (float*)d_out;

    const int n  = in_sizes[0] / FIN;
    const int E  = in_sizes[1] / 2;
    const int ET = E + n;

    // carve workspace
    char* wsp = (char*)d_ws;
    auto carve = [&](size_t bytes) { void* p = (void*)wsp; wsp += (bytes + 255) & ~(size_t)255; return p; };
    float* h     = (float*)carve((size_t)n * HID * 4);
    float* bufA  = (float*)carve((size_t)n * HID * 4);
    float* bufB  = (float*)carve((size_t)n * HID * 4);
    float* es    = (float*)carve((size_t)n * 4);
    float* ed    = (float*)carve((size_t)n * 4);
    float* mmax  = (float*)carve((size_t)n * 4);
    float* denom = (float*)carve((size_t)n * 4);
    float* ebuf  = (float*)carve((size_t)ET * 4);
    float* sums  = (float*)carve((size_t)NGR * HID * 4);
    float* cnt   = (float*)carve((size_t)NGR * 4);

    const int T = 256;
    const int bn64  = (n * HID + T - 1) / T;       // n*HID threads
    const int bwave = ((n + 15) / 16 + 7) / 8;     // one wave / 16 rows, 8 waves / block
    const int bnode = (n + 7) / 8;                 // one wave / node
    const int bedge = (ET + T - 1) / T;            // one thread / edge
    const int bew   = (ET + 7) / 8;                // one wave / edge

    const float* layer_in = x;
    float* outs[3] = { bufA, bufB, bufA };
    int Ks[3] = { FIN, HID, HID };

    for (int L = 0; L < 3; ++L) {
        float* agg = outs[L];
        k_init      <<<bn64,  T, 0, stream>>>(agg, mmax, denom, n);
        k_gemm_wmma <<<bwave, T, 0, stream>>>(layer_in, Wp[L], h, n, Ks[L]);
        k_score     <<<bnode, T, 0, stream>>>(h, asp[L], adp[L], es, ed, n);
        k_edge_max  <<<bedge, T, 0, stream>>>(ei, E, n, es, ed, ebuf, mmax);
        k_edge_expsum<<<bedge,T, 0, stream>>>(ei, E, n, ebuf, mmax, denom);
        k_edge_agg  <<<bew,   T, 0, stream>>>(ei, E, n, ebuf, denom, h, agg);
        k_bias_relu <<<bn64,  T, 0, stream>>>(agg, bp[L], n);
        layer_in = agg;
    }

    k_pool_init <<<(NGR * HID + T - 1) / T, T, 0, stream>>>(sums, cnt);
    k_pool_acc  <<<bnode, T, 0, stream>>>(layer_in, batch, sums, cnt, n);
    k_fc        <<<(NGR * LAT + T - 1) / T, T, 0, stream>>>(sums, cnt, fcW, fcb, out);
}